// RatioFeatureDiscovery_28260884807825
// MI455X (gfx1250) — compile-verified
//
#include <hip/hip_runtime.h>
#include <math.h>

typedef float v2f __attribute__((ext_vector_type(2)));
typedef float v8f __attribute__((ext_vector_type(8)));
typedef unsigned int u32x4 __attribute__((ext_vector_type(4)));
typedef int i32x4 __attribute__((ext_vector_type(4)));
typedef int i32x8 __attribute__((ext_vector_type(8)));

#define NF 256
#define NH 128
#define NB 2048
#define KTOP 15
#define NPAIR 65536
#define NSPLIT 32            // column splits for K2
#define ROWS_PER_WG 128
#define COLS_PER_WG 2048
#define TILES_PER_WG 128     // COLS_PER_WG / 16
#define NWG_K2 512           // (NB/ROWS_PER_WG) * NSPLIT = 16*32

// workspace layout (units: floats / ints, both 4 bytes)
#define OFF_HRS   0u
#define OFF_TF    (OFF_HRS + (unsigned)(NB*NH))          // 262144
#define OFF_OPW   (OFF_TF + (unsigned)(NB*NF))           // 786432
#define OFF_ENT   (OFF_OPW + (unsigned)(NB*4))           // 794624 (512 used)
#define OFF_TOPV  (OFF_ENT + 512u)                       // 795136
#define OFF_LV    (OFF_TOPV + (unsigned)(NB*KTOP))       // 825856
#define OFF_LI    (OFF_LV + (unsigned)(NB*NSPLIT*KTOP))  // 1808896
// total floats = 1808896 + 983040 = 2791936  (~11.2 MB)

__device__ __forceinline__ v8f wmma_f32(v2f a, v2f b, v8f c) {
  // V_WMMA_F32_16X16X4_F32 : D = A(16x4) * B(4x16) + C(16x16), all fp32
  return __builtin_amdgcn_wmma_f32_16x16x4_f32(false, a, false, b, (short)0, c,
                                               false, false);
}

// TDM: load a 128(K) x 16(N) fp32 tile of rs_w2 into LDS (row stride 64B).
// 2-D tensor descriptor: tensor_dim0 = 65536 (row width), dim0_stride = 65536,
// tile_dim0 = 16 (contiguous), tile_dim1 = 128 rows. data_size = 4 bytes.
__device__ __forceinline__ void tdm_load_b_tile(const float* gptr,
                                                unsigned int lds_off) {
  const unsigned long long ga = (unsigned long long)(uintptr_t)gptr;
  u32x4 g0;
  g0[0] = 1u;                                         // count=1, user mode
  g0[1] = lds_off;                                    // LDS byte offset
  g0[2] = (unsigned int)(ga & 0xffffffffull);         // global_addr[31:0]
  g0[3] = (unsigned int)((ga >> 32) & 0x1ffffffull)   // global_addr[56:32]
          | (2u << 30);                               // type = 2 ("image")
  i32x8 g1;
  g1[0] = 0x00020000;    // data_size=2 (4B); no multicast/pad/iterate
  g1[1] = 0x00000000;    // abar=0, tensor_dim0[15:0]=0      (65536 lo)
  g1[2] = 0x00800001;    // tensor_dim0[31:16]=1, tensor_dim1[15:0]=128
  g1[3] = 0x00100000;    // tensor_dim1[31:16]=0, tile_dim0=16
  g1[4] = 0x00000080;    // tile_dim1=128, tile_dim2=0
  g1[5] = 0x00010000;    // tensor_dim0_stride[31:0] = 65536
  g1[6] = 0x00000000;    // stride hi / dim1_stride lo
  g1[7] = 0x00000000;
  i32x4 gz4 = {0, 0, 0, 0};
#if defined(__clang_major__) && (__clang_major__ >= 23)
  i32x8 gz8 = {0, 0, 0, 0, 0, 0, 0, 0};
  __builtin_amdgcn_tensor_load_to_lds(g0, g1, gz4, gz4, gz8, 0);
#else
  __builtin_amdgcn_tensor_load_to_lds(g0, g1, gz4, gz4, 0);
#endif
}

// -------------------------------------------------------------------------
// K1: per 16-row block: h_rs = relu(x@rs_w1+b), h_ft = relu(x@ft_w1+b),
//     h_os = relu(x@os_w1+b), tf = h_ft@ft_w2 + b, op_w = softmax(h_os@os_w2+b)
// -------------------------------------------------------------------------
__global__ __launch_bounds__(256) void k1_small_mlps(
    const float* __restrict__ x,
    const float* __restrict__ rs_w1, const float* __restrict__ rs_b1,
    const float* __restrict__ os_w1, const float* __restrict__ os_b1,
    const float* __restrict__ os_w2, const float* __restrict__ os_b2,
    const float* __restrict__ ft_w1, const float* __restrict__ ft_b1,
    const float* __restrict__ ft_w2, const float* __restrict__ ft_b2,
    float* __restrict__ ws)
{
  __shared__ float xs[16 * NF];     // 16 KB
  __shared__ float hft[16 * NH];    // 8 KB
  __shared__ float hos[16 * NH];    // 8 KB
  __shared__ float oplog[16 * 4];

  const int tid  = threadIdx.x;
  const int lane = tid & 31;
  const int w    = tid >> 5;        // wave 0..7
  const int half = lane >> 4;
  const int l16  = lane & 15;
  const int row0 = blockIdx.x * 16;

  for (int i = tid; i < 16 * NF; i += 256)
    xs[i] = x[row0 * NF + i];
  __syncthreads();

  // ---- layer 1: 3 weight matrices x 8 column tiles of 16 -> 24 tiles, 8 waves
  for (int s = 0; s < 3; ++s) {
    const int tileId = w + 8 * s;
    const int m  = tileId >> 3;           // 0=rs, 1=ft, 2=os
    const int c0 = (tileId & 7) * 16;
    const float* w1 = (m == 0) ? rs_w1 : (m == 1 ? ft_w1 : os_w1);
    const float* b1 = (m == 0) ? rs_b1 : (m == 1 ? ft_b1 : os_b1);
    v8f acc = {0.f,0.f,0.f,0.f,0.f,0.f,0.f,0.f};
#pragma unroll 4
    for (int kk = 0; kk < NF; kk += 4) {
      v2f a, b;
      a.x = xs[l16 * NF + kk + 2 * half];
      a.y = xs[l16 * NF + kk + 1 + 2 * half];
      b.x = w1[(kk + 2 * half) * NH + c0 + l16];
      b.y = w1[(kk + 1 + 2 * half) * NH + c0 + l16];
      acc = wmma_f32(a, b, acc);
    }
    const float bias = b1[c0 + l16];
#pragma unroll
    for (int r = 0; r < 8; ++r) {
      const int rr = r + 8 * half;
      float v = acc[r] + bias;
      v = v > 0.f ? v : 0.f;                       // relu
      if (m == 0)      ws[OFF_HRS + (size_t)(row0 + rr) * NH + c0 + l16] = v;
      else if (m == 1) hft[rr * NH + c0 + l16] = v;
      else             hos[rr * NH + c0 + l16] = v;
    }
  }
  __syncthreads();

  // ---- tf = h_ft @ ft_w2 + ft_b2 : 16x256 out, K=128 -> 16 tiles, 2 per wave
  for (int s = 0; s < 2; ++s) {
    const int c0 = (w + 8 * s) * 16;
    v8f acc = {0.f,0.f,0.f,0.f,0.f,0.f,0.f,0.f};
#pragma unroll 4
    for (int kk = 0; kk < NH; kk += 4) {
      v2f a, b;
      a.x = hft[l16 * NH + kk + 2 * half];
      a.y = hft[l16 * NH + kk + 1 + 2 * half];
      b.x = ft_w2[(kk + 2 * half) * NF + c0 + l16];
      b.y = ft_w2[(kk + 1 + 2 * half) * NF + c0 + l16];
      acc = wmma_f32(a, b, acc);
    }
    const float bias = ft_b2[c0 + l16];
#pragma unroll
    for (int r = 0; r < 8; ++r) {
      const int rr = r + 8 * half;
      ws[OFF_TF + (size_t)(row0 + rr) * NF + c0 + l16] = acc[r] + bias;
    }
  }

  // ---- op logits: 16 rows x 4 outputs, K=128 (tiny; scalar dot)
  if (tid < 64) {
    const int r = tid >> 2, o = tid & 3;
    float acc = os_b2[o];
    for (int k = 0; k < NH; ++k) acc += hos[r * NH + k] * os_w2[k * 4 + o];
    oplog[r * 4 + o] = acc;
  }
  __syncthreads();
  if (tid < 16) {
    const float a0 = oplog[tid*4+0], a1 = oplog[tid*4+1];
    const float a2 = oplog[tid*4+2], a3 = oplog[tid*4+3];
    const float mx = fmaxf(fmaxf(a0,a1), fmaxf(a2,a3));
    const float e0 = __expf(a0-mx), e1 = __expf(a1-mx);
    const float e2 = __expf(a2-mx), e3 = __expf(a3-mx);
    const float inv = 1.0f / (e0+e1+e2+e3);
    float* o = ws + OFF_OPW + (size_t)(row0 + tid) * 4;
    o[0] = e0*inv; o[1] = e1*inv; o[2] = e2*inv; o[3] = e3*inv;
  }
}

// -------------------------------------------------------------------------
// K2: fused big GEMM (h_rs @ rs_w2) -> sigmoid -> entropy + streaming top-15.
// Workgroup = 128 batch rows x 2048 columns; one shared double-buffered
// 128x16 B tile per column tile, loaded by the Tensor Data Mover; 8 waves
// each own 16 exclusive rows (private top-15 lists, no intra-WG merge).
// -------------------------------------------------------------------------
__global__ __launch_bounds__(256) void k2_big(
    const float* __restrict__ rs_w2, const float* __restrict__ rs_b2,
    float* __restrict__ ws)
{
  __shared__ float hs[ROWS_PER_WG * NH];        // 64 KB: h_rs rows
  __shared__ float btile[2][NH * 16];           // 16 KB: double-buffered B
  __shared__ float lval[8][16][KTOP];           // 7.5 KB
  __shared__ int   lidx[8][16][KTOP];           // 7.5 KB
  volatile __shared__ float stv[8][32];
  volatile __shared__ int   sti[8][32];
  __shared__ float entp[8];

  const int tid  = threadIdx.x;
  const int lane = tid & 31;
  const int w    = tid >> 5;
  const int half = lane >> 4;
  const int l16  = lane & 15;
  const int bid  = blockIdx.x;
  const int rowgrp = bid >> 5;                  // 0..15
  const int cs     = bid & 31;                  // 0..31 column split
  const int row0   = rowgrp * ROWS_PER_WG;
  const int colbase = cs * COLS_PER_WG;

  // load 128 rows of h_rs (vectorized), init lists
  {
    const float4* src = (const float4*)(ws + OFF_HRS + (size_t)row0 * NH);
    float4* dst = (float4*)hs;
    for (int i = tid; i < ROWS_PER_WG * NH / 4; i += 256) dst[i] = src[i];
  }
  for (int i = tid; i < 8 * 16 * KTOP; i += 256) {
    (&lval[0][0][0])[i] = -3.4e38f;
    (&lidx[0][0][0])[i] = 0x7fffffff;
  }
  if (tid < 8) entp[tid] = 0.f;

  // prologue: wave 0 kicks off TDM for tile 0
  if (w == 0 && lane == 0)
    tdm_load_b_tile(rs_w2 + colbase,
                    (unsigned int)(uintptr_t)&btile[0][0]);

  float elocal = 0.f;

  for (int t = 0; t < TILES_PER_WG; ++t) {
    const int col0 = colbase + t * 16;
    if (w == 0 && lane == 0) {
      if (t + 1 < TILES_PER_WG)
        tdm_load_b_tile(rs_w2 + colbase + (t + 1) * 16,
                        (unsigned int)(uintptr_t)&btile[(t + 1) & 1][0]);
    }
    if (w == 0) {
      if (t + 1 < TILES_PER_WG)
        __builtin_amdgcn_s_wait_tensorcnt(1);   // tile t done, t+1 in flight
      else
        __builtin_amdgcn_s_wait_tensorcnt(0);   // last tile done
    }
    __syncthreads();                 // (A) btile[t&1] ready for everyone

    const float* bt = &btile[t & 1][0];
    v8f acc = {0.f,0.f,0.f,0.f,0.f,0.f,0.f,0.f};
#pragma unroll 8
    for (int kk = 0; kk < NH; kk += 4) {
      v2f a, b;
      a.x = hs[(w * 16 + l16) * NH + kk + 2 * half];
      a.y = hs[(w * 16 + l16) * NH + kk + 1 + 2 * half];
      b.x = bt[(kk + 2 * half) * 16 + l16];
      b.y = bt[(kk + 1 + 2 * half) * 16 + l16];
      acc = wmma_f32(a, b, acc);
    }
    const float bias = rs_b2[col0 + l16];

#pragma unroll
    for (int r = 0; r < 8; ++r) {
      const float z   = acc[r] + bias;
      const float sel = 1.0f / (1.0f + __expf(-z));
      elocal += sel * __logf(sel + 1e-8f);

      const int rr   = r + 8 * half;          // row within this wave's 16
      const int jidx = col0 + l16;            // global pair index
      float m16 = sel;                        // 16-lane tile max gate
      m16 = fmaxf(m16, __shfl_xor(m16, 1, 16));
      m16 = fmaxf(m16, __shfl_xor(m16, 2, 16));
      m16 = fmaxf(m16, __shfl_xor(m16, 4, 16));
      m16 = fmaxf(m16, __shfl_xor(m16, 8, 16));
      const float kth = lval[w][rr][KTOP - 1];
      if (m16 > kth) {                        // uniform per 16-lane group
        stv[w][lane] = sel;
        sti[w][lane] = jidx;
        if (l16 == 0) {                       // leader serial insert
#pragma unroll 1
          for (int i = 0; i < 16; ++i) {
            const float v  = stv[w][half * 16 + i];
            const int   jx = sti[w][half * 16 + i];
            if (v > lval[w][rr][KTOP - 1]) {
              lval[w][rr][KTOP - 1] = v;
              lidx[w][rr][KTOP - 1] = jx;
#pragma unroll 1
              for (int p = KTOP - 1; p > 0; --p) {
                const float va = lval[w][rr][p], vb = lval[w][rr][p - 1];
                const int   ia = lidx[w][rr][p], ib = lidx[w][rr][p - 1];
                if (va > vb || (va == vb && ia < ib)) {
                  lval[w][rr][p] = vb; lval[w][rr][p - 1] = va;
                  lidx[w][rr][p] = ib; lidx[w][rr][p - 1] = ia;
                } else break;
              }
            }
          }
        }
      }
    }
    __syncthreads();                 // (B) everyone done with btile[t&1]
  }

  // wave entropy reduction
  elocal += __shfl_xor(elocal, 16, 32);
  elocal += __shfl_xor(elocal,  8, 32);
  elocal += __shfl_xor(elocal,  4, 32);
  elocal += __shfl_xor(elocal,  2, 32);
  elocal += __shfl_xor(elocal,  1, 32);
  if (lane == 0) entp[w] = elocal;

  // write this wave's 16 row-lists (exclusive rows, no merge needed)
  for (int i = lane; i < 16 * KTOP; i += 32) {
    const int r = i / KTOP, k = i % KTOP;
    const int grow = row0 + w * 16 + r;
    ws[OFF_LV + ((size_t)grow * NSPLIT + cs) * KTOP + k] = lval[w][r][k];
    ((int*)ws)[OFF_LI + ((size_t)grow * NSPLIT + cs) * KTOP + k] = lidx[w][r][k];
  }
  __syncthreads();
  if (tid == 0) {
    float s = 0.f;
    for (int i = 0; i < 8; ++i) s += entp[i];
    ws[OFF_ENT + bid] = s;
  }
}

// -------------------------------------------------------------------------
// K3: per-row merge of 32 split lists -> final top-15; gather tf, ratio ops
// -------------------------------------------------------------------------
__global__ __launch_bounds__(256) void k3_rows(
    float* __restrict__ ws, float* __restrict__ out)
{
  const int b = blockIdx.x * 256 + threadIdx.x;   // 0..2047
  const float* tf  = ws + OFF_TF + (size_t)b * NF;
  const float* opw = ws + OFF_OPW + (size_t)b * 4;

  float val[KTOP]; int idx[KTOP];
#pragma unroll
  for (int k = 0; k < KTOP; ++k) { val[k] = -3.4e38f; idx[k] = 0x7fffffff; }

  for (int s = 0; s < NSPLIT; ++s) {
    const float* lv = ws + OFF_LV + ((size_t)b * NSPLIT + s) * KTOP;
    const int*   li = (const int*)ws + OFF_LI + ((size_t)b * NSPLIT + s) * KTOP;
    for (int k = 0; k < KTOP; ++k) {
      const float v  = lv[k];
      const int   jx = li[k];
      if (v < val[KTOP - 1]) break;              // list sorted desc
      if (v > val[KTOP - 1] || (v == val[KTOP - 1] && jx < idx[KTOP - 1])) {
        val[KTOP - 1] = v; idx[KTOP - 1] = jx;
#pragma unroll
        for (int p = KTOP - 1; p > 0; --p) {
          const bool sw = val[p] > val[p - 1] ||
                          (val[p] == val[p - 1] && idx[p] < idx[p - 1]);
          if (sw) {
            float tv = val[p]; val[p] = val[p - 1]; val[p - 1] = tv;
            int   ti = idx[p]; idx[p] = idx[p - 1]; idx[p - 1] = ti;
          }
        }
      }
    }
  }

  const float w0 = opw[0], w1 = opw[1], w2 = opw[2], w3 = opw[3];
#pragma unroll
  for (int k = 0; k < KTOP; ++k) {
    const int ij = idx[k];
    const int i2 = ij >> 8, j2 = ij & 255;       // nf = 256
    const float fi = tf[i2], fj = tf[j2];
    const float afj = fabsf(fj) + 1e-8f;
    const float ratio = fi / afj;
    const float lr = __logf(fabsf(fi) + 1e-8f) - __logf(afj);
    const float rt = ratio * w0 + lr * w1 + (fi - fj) * w2 + fi * fj * w3;
    out[(size_t)b * KTOP + k] = rt;
    ws[OFF_TOPV + (size_t)b * KTOP + k] = val[k];
  }
}

// -------------------------------------------------------------------------
// K4: deterministic fixed-order reductions for the small outputs
// -------------------------------------------------------------------------
__global__ __launch_bounds__(64) void k4_final(
    const float* __restrict__ ws, float* __restrict__ out)
{
  const int t = threadIdx.x;
  const int base = NB * KTOP;                    // 30720
  if (t < KTOP) {                                // pair_importances
    double s = 0.0;
    for (int b = 0; b < NB; ++b) s += ws[OFF_TOPV + (size_t)b * KTOP + t];
    out[base + t] = (float)(s / NB);
  } else if (t >= 16 && t < 20) {                // op_prefs
    const int o = t - 16;
    double s = 0.0;
    for (int b = 0; b < NB; ++b) s += ws[OFF_OPW + (size_t)b * 4 + o];
    out[base + KTOP + o] = (float)(s / NB);
  } else if (t >= 32 && t < 32 + KTOP) {         // ratio_mags
    const int k = t - 32;
    double s = 0.0;
    for (int b = 0; b < NB; ++b) s += fabsf(out[(size_t)b * KTOP + k]);
    out[base + KTOP + 4 + k] = (float)(s / NB);
  } else if (t == 63) {                          // sel_entropy
    double s = 0.0;
    for (int i = 0; i < NWG_K2; ++i) s += ws[OFF_ENT + i];
    out[base + KTOP + 4 + KTOP] = (float)(-s / NB);
  }
}

extern "C" void kernel_launch(void* const* d_in, const int* in_sizes, int n_in,
                              void* d_out, int out_size, void* d_ws, size_t ws_size,
                              hipStream_t stream)
{
  (void)in_sizes; (void)n_in; (void)out_size; (void)ws_size;
  const float* x     = (const float*)d_in[0];
  const float* rs_w1 = (const float*)d_in[1];
  const float* rs_b1 = (const float*)d_in[2];
  const float* rs_w2 = (const float*)d_in[3];
  const float* rs_b2 = (const float*)d_in[4];
  const float* os_w1 = (const float*)d_in[5];
  const float* os_b1 = (const float*)d_in[6];
  const float* os_w2 = (const float*)d_in[7];
  const float* os_b2 = (const float*)d_in[8];
  const float* ft_w1 = (const float*)d_in[9];
  const float* ft_b1 = (const float*)d_in[10];
  const float* ft_w2 = (const float*)d_in[11];
  const float* ft_b2 = (const float*)d_in[12];
  float* ws  = (float*)d_ws;
  float* out = (float*)d_out;

  hipLaunchKernelGGL(k1_small_mlps, dim3(NB / 16), dim3(256), 0, stream,
                     x, rs_w1, rs_b1, os_w1, os_b1, os_w2, os_b2,
                     ft_w1, ft_b1, ft_w2, ft_b2, ws);
  hipLaunchKernelGGL(k2_big, dim3(NWG_K2), dim3(256), 0, stream,
                     rs_w2, rs_b2, ws);
  hipLaunchKernelGGL(k3_rows, dim3(NB / 256), dim3(256), 0, stream, ws, out);
  hipLaunchKernelGGL(k4_final, dim3(1), dim3(64), 0, stream, ws, out);
}